// Word2Vec_40072044871828
// MI455X (gfx1250) — compile-verified
//
#include <hip/hip_runtime.h>

// -------- problem constants (match reference) --------
#define VOCAB  100000
#define DIM    128
#define BATCH  262144
#define NEG    10
#define NTGT   (NEG + 1)     // 1 positive + 10 negatives
#define EPSF   1e-9f

typedef __attribute__((ext_vector_type(2))) float v2f;   // A/B frag: 16x4 / 4x16 f32 = 2 VGPRs
typedef __attribute__((ext_vector_type(8))) float v8f;   // C/D: 16x16 f32 = 8 VGPRs

__global__ void w2v_zero_out(float* out) { out[0] = 0.0f; }

// One wave32 handles one group of 16 batch elements.
// D = A(16x4 center rows) x B(4x16 target rows, transposed) accumulated over 32 K-steps;
// diag(D) = the 16 wanted dot products, full f32 precision via V_WMMA_F32_16X16X4_F32.
__global__ __launch_bounds__(256) void w2v_loss_kernel(
    const int*   __restrict__ center,
    const int*   __restrict__ context,
    const int*   __restrict__ negatives,
    const float* __restrict__ W_in,
    const float* __restrict__ W_out,
    float*       __restrict__ out)
{
    const int lane  = threadIdx.x & 31;
    const int wave  = threadIdx.x >> 5;                 // 0..7
    const int group = blockIdx.x * 8 + wave;            // 16 batch elements per group
    const int b0    = group * 16;

    const int e  = lane & 15;   // batch element (row of A / column of B) owned by this lane
    const int hf = lane >> 4;   // 0 -> K-pair {0,1}, 1 -> K-pair {2,3} within each 4-wide K step

    // --- per-lane base pointers into the embedding tables (L2-resident gathers) ---
    const int    cidx  = center[b0 + e];
    const float* aBase = W_in + (size_t)cidx * DIM + 2 * hf;

    const float* bBase[NTGT];
    {
        const int t0 = context[b0 + e];
        bBase[0] = W_out + (size_t)t0 * DIM + 2 * hf;
#pragma unroll
        for (int k = 0; k < NEG; ++k) {
            const int tk = negatives[(b0 + e) * NEG + k];
            bBase[k + 1] = W_out + (size_t)tk * DIM + 2 * hf;
        }
    }

    // --- 11 independent f32 accumulators: A frag loaded once per K-step, reused 11x ---
    v8f acc[NTGT];
#pragma unroll
    for (int t = 0; t < NTGT; ++t) acc[t] = (v8f){0,0,0,0,0,0,0,0};

#pragma unroll 4
    for (int j = 0; j < DIM / 4; ++j) {                 // 32 K-steps of 4
        const v2f a = *(const v2f*)(aBase + 4 * j);
#pragma unroll
        for (int t = 0; t < NTGT; ++t) {
            const v2f b = *(const v2f*)(bBase[t] + 4 * j);
            acc[t] = __builtin_amdgcn_wmma_f32_16x16x4_f32(
                /*neg_a=*/false, a, /*neg_b=*/false, b,
                /*c_mod=*/(short)0, acc[t], /*reuse_a=*/false, /*reuse_b=*/false);
        }
    }

    // --- diagonal extraction + loss terms (branch-free; EXEC stays all-1s) ---
    // D layout: VGPR r, lanes 0-15 -> (M=r, N=lane); lanes 16-31 -> (M=r+8, N=lane-16).
    // Diagonal (i,i): i<8 on lane i comp i; i>=8 on lane i+16 comp i-8.
    const int  rsel = e - 8 * hf;
    const bool has  = (rsel >= 0) && (rsel < 8);

    float local = 0.0f;
#pragma unroll
    for (int t = 0; t < NTGT; ++t) {
        float s = 0.0f;
#pragma unroll
        for (int r = 0; r < 8; ++r) s = (rsel == r) ? acc[t][r] : s;
        if (t > 0) s = -s;                               // negatives use sigmoid(-score)
        const float sig  = 1.0f / (1.0f + __expf(-s));
        const float term = __logf(sig + EPSF);
        local += has ? term : 0.0f;
    }

    // --- wave32 reduction ---
#pragma unroll
    for (int off = 16; off >= 1; off >>= 1)
        local += __shfl_xor(local, off, 32);

    // --- block reduction (8 waves) + one global f32 atomic per block ---
    __shared__ float red[8];
    if (lane == 0) red[wave] = local;
    __syncthreads();
    if (threadIdx.x == 0) {
        float sum = 0.0f;
#pragma unroll
        for (int w = 0; w < 8; ++w) sum += red[w];
        atomicAdd(out, sum * (-1.0f / (float)BATCH));    // loss = -(1/B) * sum of log terms
    }
}

extern "C" void kernel_launch(void* const* d_in, const int* in_sizes, int n_in,
                              void* d_out, int out_size, void* d_ws, size_t ws_size,
                              hipStream_t stream) {
    (void)in_sizes; (void)n_in; (void)out_size; (void)d_ws; (void)ws_size;
    const int*   center    = (const int*)d_in[0];
    const int*   context   = (const int*)d_in[1];
    const int*   negatives = (const int*)d_in[2];
    const float* W_in      = (const float*)d_in[3];
    const float* W_out     = (const float*)d_in[4];
    float*       out       = (float*)d_out;

    w2v_zero_out<<<1, 1, 0, stream>>>(out);

    // BATCH/16 = 16384 groups; 8 waves (groups) per 256-thread block -> 2048 blocks.
    w2v_loss_kernel<<<BATCH / 16 / 8, 256, 0, stream>>>(
        center, context, negatives, W_in, W_out, out);
}